// DotAttentionEluX_72378788872880
// MI455X (gfx1250) — compile-verified
//
#include <hip/hip_runtime.h>

// Linear-attention (ELU+1) collapse:
//   out[b,h,i,:] = (sum_j ks[j]*V[j,:]) / (sum_j ks[j]),  ks[j]=sum_d elu(K[j,d])+1
// Query is mathematically irrelevant after row-normalization (rank-1 scores).
// Bandwidth-bound: 48MB total traffic -> ~2.1us at 23.3 TB/s.

typedef __attribute__((ext_vector_type(2))) float v2f;
typedef __attribute__((ext_vector_type(8))) float v8f;

#define SEQ_L 2048
#define DIM   64
#define CHUNK 128
#define NCH   (SEQ_L / CHUNK)   // 16

__device__ __forceinline__ float elu1(float x) {
  // jax.nn.elu(x) + 1 == (x > 0 ? x + 1 : exp(x))
  return x > 0.0f ? x + 1.0f : __expf(x);
}

// Phase 1: per (bh, chunk of 128 rows) compute partial W[64] = sum_j ks[j]*V[j,:]
// and partial S = sum_j ks[j]; write 65 floats to workspace. No atomics.
__global__ __launch_bounds__(256) void la_phase1(const float* __restrict__ Kmat,
                                                 const float* __restrict__ Vmat,
                                                 float* __restrict__ ws) {
  const int bh   = blockIdx.x;   // 0..BH-1
  const int ch   = blockIdx.y;   // 0..NCH-1
  const int tid  = threadIdx.x;  // 0..255 (8 waves of 32)
  const int wave = tid >> 5;
  const int lane = tid & 31;

  __shared__ float ksum[CHUNK];     // ks for the 128 rows of this chunk
  __shared__ float wpart[8][16];    // per-wave 16-column partial of W
  __shared__ float Sred;

  const size_t base = (size_t)bh * SEQ_L * DIM + (size_t)ch * CHUNK * DIM;
  const float* Kc = Kmat + base;
  const float* Vc = Vmat + base;

  // ---- Step A: ks[j] for 128 rows; wave w owns rows 16w..16w+15 ----
  // Per row: 32 lanes read cols [lane] and [lane+32] (two 128B coalesced loads),
  // then a 5-step wave32 xor-shuffle reduction.
  for (int r = 0; r < 16; ++r) {
    const int row = wave * 16 + r;
    float s = elu1(Kc[row * DIM + lane]) + elu1(Kc[row * DIM + lane + 32]);
    #pragma unroll
    for (int off = 16; off >= 1; off >>= 1) s += __shfl_xor(s, off, 32);
    if (lane == 0) ksum[row] = s;
  }
  __syncthreads();

  // ---- Step B: W[n] = sum_j ks[j]*V[j,n] via V_WMMA_F32_16X16X4_F32 ----
  // wave -> (column chunk nc = wave&3, row half jh = wave>>2; 64 rows each).
  // A (16x4, MxK): A[m,k] = ks[jb+k] for all m  (replicated rows)
  //   lane layout: VGPR0 = K={0|2}, VGPR1 = K={1|3}; lanes>=16 take K+2.
  // B (4x16, KxN): B[k,n] = V[jb+k, n0+n]; same K-split across lane halves.
  // => C[m,n] = sum_k ks[jb+k]*V[jb+k, n0+n], identical for every m.
  const int nc  = wave & 3;
  const int jh  = wave >> 2;
  const int n   = nc * 16 + (lane & 15);
  const int khi = (lane >> 4) << 1;   // 0 for lanes 0-15, 2 for lanes 16-31

  v8f c = {0.f, 0.f, 0.f, 0.f, 0.f, 0.f, 0.f, 0.f};
  #pragma unroll 4
  for (int jc = 0; jc < 16; ++jc) {
    const int jb = jh * 64 + jc * 4 + khi;
    v2f a, b;
    a[0] = ksum[jb];                 // LDS broadcast (no bank conflicts)
    a[1] = ksum[jb + 1];
    b[0] = Vc[jb * DIM + n];         // two 64B segments per load instruction
    b[1] = Vc[(jb + 1) * DIM + n];
    c = __builtin_amdgcn_wmma_f32_16x16x4_f32(
        /*neg_a=*/false, a, /*neg_b=*/false, b,
        /*c_mod=*/(short)0, c, /*reuse_a=*/false, /*reuse_b=*/false);
  }
  // Row M=0 lives in c[0] of lanes 0..15 (N = lane).
  if (lane < 16) wpart[wave][lane] = c[0];

  // ---- partial S = sum of the 128 ks values (wave 0) ----
  if (wave == 0) {
    float s = ksum[lane] + ksum[lane + 32] + ksum[lane + 64] + ksum[lane + 96];
    #pragma unroll
    for (int off = 16; off >= 1; off >>= 1) s += __shfl_xor(s, off, 32);
    if (lane == 0) Sred = s;
  }
  __syncthreads();

  float* o = ws + (size_t)(bh * NCH + ch) * (DIM + 1);
  if (tid < DIM) {
    const int cc = tid >> 4, nn = tid & 15;      // combine the two row-halves
    o[tid] = wpart[cc][nn] + wpart[cc + 4][nn];
  }
  if (tid == DIM) o[DIM] = Sred;
}

// Phase 2: reduce the 16 chunk partials (tiny) and broadcast-store 128 output
// rows per block with b128 stores.
__global__ __launch_bounds__(256) void la_phase2(const float* __restrict__ ws,
                                                 float* __restrict__ out) {
  const int bh  = blockIdx.x;
  const int ch  = blockIdx.y;
  const int tid = threadIdx.x;

  __shared__ float wsum[DIM];
  __shared__ float Ssh;

  if (tid < DIM) {
    float w = 0.f;
    #pragma unroll
    for (int cc = 0; cc < NCH; ++cc)
      w += ws[(size_t)(bh * NCH + cc) * (DIM + 1) + tid];
    wsum[tid] = w;
  } else if (tid == DIM) {
    float s = 0.f;
    #pragma unroll
    for (int cc = 0; cc < NCH; ++cc)
      s += ws[(size_t)(bh * NCH + cc) * (DIM + 1) + DIM];
    Ssh = s;
  }
  __syncthreads();

  const float inv = 1.0f / Ssh;
  const int d0 = (tid * 4) & (DIM - 1);
  float4 v;
  v.x = wsum[d0 + 0] * inv;
  v.y = wsum[d0 + 1] * inv;
  v.z = wsum[d0 + 2] * inv;
  v.w = wsum[d0 + 3] * inv;

  float4* op = (float4*)(out + (size_t)bh * SEQ_L * DIM + (size_t)ch * CHUNK * DIM);
  #pragma unroll
  for (int i = 0; i < (CHUNK * DIM) / (4 * 256); ++i)   // 8 b128 stores/thread
    op[i * 256 + tid] = v;
}

extern "C" void kernel_launch(void* const* d_in, const int* in_sizes, int n_in,
                              void* d_out, int out_size, void* d_ws, size_t ws_size,
                              hipStream_t stream) {
  // inputs: query (unused by the math), key, value — all float32 [B,H,L,D]
  const float* K = (const float*)d_in[1];
  const float* V = (const float*)d_in[2];
  float* out = (float*)d_out;
  float* ws  = (float*)d_ws;      // needs BH*NCH*65*4 = 130 KB

  const int BH = in_sizes[1] / (SEQ_L * DIM);   // 32
  dim3 grid((unsigned)BH, NCH);
  la_phase1<<<grid, 256, 0, stream>>>(K, V, ws);
  la_phase2<<<grid, 256, 0, stream>>>(ws, out);
  (void)d_in; (void)n_in; (void)out_size; (void)ws_size;
}